// HVIMoGP_rBCM_44547400794494
// MI455X (gfx1250) — compile-verified
//
#include <hip/hip_runtime.h>
#include <hip/hip_bf16.h>

// ---------------------------------------------------------------------------
// HVI MoGP rBCM for MI455X (gfx1250, wave32, WMMA)
//
//  Kernel 1: route_kernel      — 2-level GMM hard routing (VALU)
//  Kernel 2: gp_factor_kernel  — per-GP (580 WGs) kernel-matrix build in LDS
//                                (256x257 fp32, 257KB of the 320KB WGP LDS),
//                                blocked Cholesky w/ v_wmma_f32_16x16x4_f32
//                                trailing updates, alpha solve, write L^T.
//                                Xtr staged via TDM tensor_load_to_lds.
//  Kernel 3: predict_kernel    — per test point: k*, mu=k*.alpha,
//                                q=||L^-1 k*||^2 (== k*^T K^-1 k*), rBCM.
//
//  Workspace: eidx/gidx (8KB) + alpha (580*1KB) + L^T (580*256KB) ~ 153MB.
// ---------------------------------------------------------------------------

#define DF     6
#define GGEO   20
#define KPHI   27
#define NEXP   540
#define MPTS   256
#define NTEST  1024
#define NGP    580      // 540 experts + 40 baselines (both 256 points)
#define JITTER 1e-4f
#define LD     257      // padded LDS leading dim: stride 257 words -> bank-conflict-free columns

typedef __attribute__((ext_vector_type(2))) float        v2f;
typedef __attribute__((ext_vector_type(8))) float        v8f;
typedef __attribute__((ext_vector_type(4))) unsigned int v4u;
typedef __attribute__((ext_vector_type(8))) int          v8i;
typedef __attribute__((ext_vector_type(4))) int          v4i;

// ---------------------------------------------------------------------------
// Kernel 1: hard 2-level GMM routing
// ---------------------------------------------------------------------------
__global__ __launch_bounds__(256) void route_kernel(
    const float* __restrict__ X_test,
    const float* __restrict__ smean, const float* __restrict__ sscale,
    const float* __restrict__ gmeans, const float* __restrict__ glogvar,
    const float* __restrict__ glogw,
    const float* __restrict__ pmeans, const float* __restrict__ plogvar,
    const float* __restrict__ plogw,
    int* __restrict__ eidx, int* __restrict__ gidx)
{
  int n = blockIdx.x * blockDim.x + threadIdx.x;
  if (n >= NTEST) return;
  const float LOG2PI = 1.8378770664093453f;
  float xs[DF];
#pragma unroll
  for (int d = 0; d < DF; ++d)
    xs[d] = (X_test[n * DF + d] - smean[d]) / sscale[d];

  int bg = 0; float best = -3.4e38f;
  for (int g = 0; g < GGEO; ++g) {
    float sl = 0.f, q = 0.f;
#pragma unroll
    for (int d = 0; d < DF; ++d) {
      float lv = glogvar[g * DF + d];
      float df = xs[d] - gmeans[g * DF + d];
      sl += lv; q += df * df * expf(-lv);
    }
    float lp = glogw[g] - 0.5f * (sl + q + DF * LOG2PI);
    if (lp > best) { best = lp; bg = g; }
  }
  int bk = 0; best = -3.4e38f;
  for (int k = 0; k < KPHI; ++k) {
    int idx = bg * KPHI + k;
    float sl = 0.f, q = 0.f;
#pragma unroll
    for (int d = 0; d < DF; ++d) {
      float lv = plogvar[idx * DF + d];
      float df = xs[d] - pmeans[idx * DF + d];
      sl += lv; q += df * df * expf(-lv);
    }
    float lp = plogw[idx] - 0.5f * (sl + q + DF * LOG2PI);
    if (lp > best) { best = lp; bk = k; }
  }
  gidx[n] = bg;
  eidx[n] = bg * KPHI + bk;
}

// ---------------------------------------------------------------------------
// Kernel 2: per-GP kernel build + blocked Cholesky (WMMA f32) + alpha solve.
// One workgroup (8 wave32) per GP; K resident in LDS.
// ---------------------------------------------------------------------------
__global__ __launch_bounds__(256) void gp_factor_kernel(
    const float* __restrict__ X_exp,  const float* __restrict__ Y_exp,
    const float* __restrict__ X_base, const float* __restrict__ Y_base,
    const float* __restrict__ lls_e, const float* __restrict__ los_e, const float* __restrict__ lno_e,
    const float* __restrict__ lls_b, const float* __restrict__ los_b, const float* __restrict__ lno_b,
    float* __restrict__ wsU, float* __restrict__ wsA)
{
  __shared__ float Ksh[MPTS * LD];   // 257 KB: K -> L in place (lower triangle)
  __shared__ float Xsh[MPTS * DF];   // 6 KB training points
  __shared__ float bsh[MPTS];
  __shared__ float zsh[MPTS];

  const int gid = blockIdx.x;
  const int tid = threadIdx.x;

  const float* Xtr; const float* Ytr; float lls, los_, lno;
  if (gid < NEXP) {
    Xtr = X_exp  + (size_t)gid * MPTS * DF; Ytr = Y_exp  + (size_t)gid * MPTS;
    lls = lls_e[gid]; los_ = los_e[gid]; lno = lno_e[gid];
  } else {
    int b = gid - NEXP;
    Xtr = X_base + (size_t)b * MPTS * DF;   Ytr = Y_base + (size_t)b * MPTS;
    lls = lls_b[b]; los_ = los_b[b]; lno = lno_b[b];
  }
  const float inv_ls2 = expf(-2.f * lls);
  const float osv     = expf(los_);
  const float nv      = expf(lno) + JITTER;

  // ---- TDM: async DMA Xtr (1536 f32, 1-D tile) into LDS; wave 0 issues ----
  if (tid < 32) {
    unsigned long long ga = (unsigned long long)(size_t)Xtr;
    unsigned int lds = (unsigned int)(size_t)(&Xsh[0]);
    v4u g0;
    g0.x = 1u;                                              // count=1, gather off
    g0.y = lds;                                             // lds_addr
    g0.z = (unsigned int)ga;                                // global_addr[31:0]
    g0.w = ((unsigned int)(ga >> 32) & 0x1FFFFFFu) | (2u << 30); // addr[56:32] | type=2
    v8i g1;
    g1.s0 = (int)(2u << 16);                 // data_size = 4B, no multicast/iter/pad
    g1.s1 = (int)((unsigned)(MPTS * DF) << 16); // tensor_dim0[15:0] (atomic_bar_addr=0)
    g1.s2 = (int)(1u << 16);                 // tensor_dim0 hi=0, tensor_dim1=1
    g1.s3 = (int)((unsigned)(MPTS * DF) << 16); // tile_dim0 = 1536
    g1.s4 = 0;                               // tile_dim1/2 unused (1-D)
    g1.s5 = (int)(MPTS * DF);                // tensor_dim0_stride
    g1.s6 = 0; g1.s7 = 0;
    v4i zz4 = {0, 0, 0, 0};
    v8i zz8 = {0, 0, 0, 0, 0, 0, 0, 0};
    __builtin_amdgcn_tensor_load_to_lds(g0, g1, zz4, zz4, zz8, 0);
    __builtin_amdgcn_s_wait_tensorcnt(0);
  }
  bsh[tid] = Ytr[tid];
  __syncthreads();

  // ---- build K = os*exp(-0.5*d2/ls2) + nv*I : one row per thread ----
  {
    float xr[DF];
#pragma unroll
    for (int d = 0; d < DF; ++d) xr[d] = Xsh[tid * DF + d];
    for (int j = 0; j < MPTS; ++j) {
      float d2 = 0.f;
#pragma unroll
      for (int d = 0; d < DF; ++d) { float df = xr[d] - Xsh[j * DF + d]; d2 += df * df; }
      float v = osv * expf(-0.5f * d2 * inv_ls2);
      Ksh[tid * LD + j] = (j == tid) ? v + nv : v;
    }
  }
  __syncthreads();

  // ---- blocked right-looking Cholesky, block size 16, SYRK via WMMA ----
  // Wave id pinned to an SGPR so the tile loop is scalar control flow and
  // EXEC provably stays all-ones around v_wmma (ISA 7.12 requirement).
  const int wave = __builtin_amdgcn_readfirstlane(tid >> 5);
  const int lane = tid & 31;
  const int lrow = lane & 15;
  const int koff = (lane < 16) ? 0 : 2;          // A/B fragment K-slot per half-wave
  const int chal = (lane < 16) ? 0 : 8;          // C fragment M-offset per half-wave

  for (int kb = 0; kb < 16; ++kb) {
    // diagonal 16x16 Cholesky (serial, tiny)
    if (tid == 0) {
      int base = kb * 16;
      for (int c = 0; c < 16; ++c) {
        int rc = base + c;
        float dsum = Ksh[rc * LD + rc];
        for (int t = 0; t < c; ++t) { float l = Ksh[rc * LD + base + t]; dsum -= l * l; }
        dsum = sqrtf(fmaxf(dsum, 1e-20f));
        Ksh[rc * LD + rc] = dsum;
        float inv = 1.f / dsum;
        for (int r = c + 1; r < 16; ++r) {
          int rr = base + r;
          float s = Ksh[rr * LD + rc];
          for (int t = 0; t < c; ++t) s -= Ksh[rr * LD + base + t] * Ksh[rc * LD + base + t];
          Ksh[rr * LD + rc] = s * inv;
        }
      }
    }
    __syncthreads();

    // panel TRSM: one thread per row below the diagonal block
    {
      int nrows = MPTS - (kb + 1) * 16;
      if (tid < nrows) {
        int r = (kb + 1) * 16 + tid, base = kb * 16;
        for (int c = 0; c < 16; ++c) {
          int cc = base + c;
          float s = Ksh[r * LD + cc];
          for (int t = 0; t < c; ++t) s -= Ksh[r * LD + base + t] * Ksh[cc * LD + base + t];
          Ksh[r * LD + cc] = s / Ksh[cc * LD + cc];
        }
      }
    }
    __syncthreads();

    // trailing SYRK: C_ij -= L_ik * L_jk^T via 4 chained v_wmma_f32_16x16x4_f32
    int rem = 15 - kb;
    int ntiles = rem * (rem + 1) / 2;
    for (int t = wave; t < ntiles; t += 8) {       // scalar (SGPR) loop: EXEC untouched
      int tt = t, bj = kb + 1;
      while (tt >= 16 - bj) { tt -= 16 - bj; ++bj; }  // scalar: tt/bj derive from SGPRs
      int bi = bj + tt;

      int m0 = bi * 16 + chal;
      int nn = bj * 16 + lrow;
      v8f c;
#pragma unroll
      for (int r = 0; r < 8; ++r) c[r] = Ksh[(m0 + r) * LD + nn];

      int am = bi * 16 + lrow;
      int bn = bj * 16 + lrow;
#pragma unroll
      for (int kk = 0; kk < 4; ++kk) {
        int kc = kb * 16 + kk * 4 + koff;
        v2f a, b;
        a.x = -Ksh[am * LD + kc];  a.y = -Ksh[am * LD + kc + 1];   // (-A)·B + C == C - A·B^T
        b.x =  Ksh[bn * LD + kc];  b.y =  Ksh[bn * LD + kc + 1];
        c = __builtin_amdgcn_wmma_f32_16x16x4_f32(false, a, false, b, (short)0, c, false, false);
      }
#pragma unroll
      for (int r = 0; r < 8; ++r) Ksh[(m0 + r) * LD + nn] = c[r];
    }
    __syncthreads();
  }

  // ---- alpha = K^-1 y : forward then backward substitution (column sweep) ----
  for (int j = 0; j < MPTS; ++j) {
    if (tid == j) zsh[j] = bsh[j] / Ksh[j * LD + j];
    __syncthreads();
    if (tid > j) bsh[tid] -= Ksh[tid * LD + j] * zsh[j];
    __syncthreads();
  }
  for (int j = MPTS - 1; j >= 0; --j) {
    if (tid == j) zsh[j] = zsh[j] / Ksh[j * LD + j];
    __syncthreads();
    if (tid < j) zsh[tid] -= Ksh[j * LD + tid] * zsh[j];
    __syncthreads();
  }

  // ---- write U = L^T (row j of U = column j of L -> coalesced solve reads) ----
  {
    float* Uo = wsU + (size_t)gid * MPTS * MPTS;
    for (int idx = tid; idx < MPTS * MPTS; idx += 256) {
      int i = idx >> 8, j = idx & 255;             // U[i][j] = L[j][i]
      Uo[idx] = (j >= i) ? Ksh[j * LD + i] : 0.f;  // LDS read stride-257 -> conflict-free
    }
    wsA[(size_t)gid * MPTS + tid] = zsh[tid];
  }
}

// ---------------------------------------------------------------------------
// Kernel 3: per-point prediction + rBCM combine. One WG per test point.
// q = ||L^-1 k*||^2 == k*^T K^-1 k*  (identical to reference's Kinv path).
// L^T rows are coalesced global reads; the whole L store (~149MB) is
// L2-resident (192MB), so the serial solve streams from L2 (+prefetch).
// ---------------------------------------------------------------------------
__global__ __launch_bounds__(256) void predict_kernel(
    const float* __restrict__ X_test,
    const float* __restrict__ X_exp, const float* __restrict__ X_base,
    const float* __restrict__ lls_e, const float* __restrict__ los_e, const float* __restrict__ lno_e,
    const float* __restrict__ lls_b, const float* __restrict__ los_b, const float* __restrict__ lno_b,
    const int* __restrict__ null_mask,
    const int* __restrict__ eidx, const int* __restrict__ gidx,
    const float* __restrict__ wsU, const float* __restrict__ wsA,
    float* __restrict__ out)
{
  __shared__ float ksh[MPTS];
  __shared__ float vsh[MPTS];
  __shared__ float red[MPTS];
  __shared__ float res[4];     // mu_e, q_e, mu_b, q_b

  const int n = blockIdx.x, tid = threadIdx.x;
  const int e = eidx[n], g = gidx[n];
  float xq[DF];
#pragma unroll
  for (int d = 0; d < DF; ++d) xq[d] = X_test[n * DF + d];

  for (int pass = 0; pass < 2; ++pass) {
    const int gi = (pass == 0) ? e : (NEXP + g);
    const float* Xtr = (pass == 0) ? (X_exp + (size_t)e * MPTS * DF)
                                   : (X_base + (size_t)g * MPTS * DF);
    const float lls  = (pass == 0) ? lls_e[e] : lls_b[g];
    const float los_ = (pass == 0) ? los_e[e] : los_b[g];
    const float inv_ls2 = expf(-2.f * lls);
    const float osv     = expf(los_);
    const float* U  = wsU + (size_t)gi * MPTS * MPTS;
    const float* al = wsA + (size_t)gi * MPTS;

    float d2 = 0.f;
#pragma unroll
    for (int d = 0; d < DF; ++d) { float df = xq[d] - Xtr[tid * DF + d]; d2 += df * df; }
    float ks = osv * expf(-0.5f * d2 * inv_ls2);
    ksh[tid] = ks;

    // mu = k* . alpha
    red[tid] = ks * al[tid];
    __syncthreads();
    for (int s = 128; s > 0; s >>= 1) { if (tid < s) red[tid] += red[tid + s]; __syncthreads(); }
    if (tid == 0) res[pass * 2 + 0] = red[0];
    __syncthreads();

    // forward solve  L v = k*   (L[i][j] == U[j][i]; row j of U coalesced)
    for (int j = 0; j < MPTS; ++j) {
      if (tid == j) vsh[j] = ksh[j] / U[j * MPTS + j];
      __syncthreads();
      if (tid > j) ksh[tid] -= U[j * MPTS + tid] * vsh[j];
      if (j + 1 < MPTS) __builtin_prefetch(&U[(j + 1) * MPTS + tid], 0, 1);
      __syncthreads();
    }
    red[tid] = vsh[tid] * vsh[tid];
    __syncthreads();
    for (int s = 128; s > 0; s >>= 1) { if (tid < s) red[tid] += red[tid + s]; __syncthreads(); }
    if (tid == 0) res[pass * 2 + 1] = red[0];
    __syncthreads();
  }

  if (tid == 0) {
    float nv_e = expf(lno_e[e]) + JITTER, nv_b = expf(lno_b[g]) + JITTER;
    float os_e = expf(los_e[e]),          os_b = expf(los_b[g]);
    float mu_e = res[0], q_e = res[1], mu_b = res[2], q_b = res[3];
    float var_e = fmaxf(os_e - q_e, JITTER) + nv_e;
    float var_b = fmaxf(os_b - q_b, JITTER) + nv_b;
    float prior_e = os_e + nv_e, prior_b = os_b + nv_b;
    float beta_e = (null_mask[e] == 0) ? 0.f
                 : fmaxf(0.5f * (logf(prior_e) - logf(var_e)), 0.f);
    float beta_b = fmaxf(0.5f * (logf(prior_b) - logf(var_b)), 0.f);
    float prec = beta_e / var_e + beta_b / var_b + (1.f - beta_e - beta_b) / prior_b;
    prec = fmaxf(prec, 1e-6f);
    out[n]         = (beta_e * mu_e / var_e + beta_b * mu_b / var_b) / prec;
    out[NTEST + n] = 1.f / prec;
  }
}

// ---------------------------------------------------------------------------
extern "C" void kernel_launch(void* const* d_in, const int* in_sizes, int n_in,
                              void* d_out, int out_size, void* d_ws, size_t ws_size,
                              hipStream_t stream) {
  (void)in_sizes; (void)n_in; (void)out_size; (void)ws_size;
  const float* X_test  = (const float*)d_in[0];
  const float* X_exp   = (const float*)d_in[1];
  const float* Y_exp   = (const float*)d_in[2];
  const float* X_base  = (const float*)d_in[3];
  const float* Y_base  = (const float*)d_in[4];
  const float* lls_e   = (const float*)d_in[5];
  const float* los_e   = (const float*)d_in[6];
  const float* lno_e   = (const float*)d_in[7];
  const float* lls_b   = (const float*)d_in[8];
  const float* los_b   = (const float*)d_in[9];
  const float* lno_b   = (const float*)d_in[10];
  const float* smean   = (const float*)d_in[11];
  const float* sscale  = (const float*)d_in[12];
  const float* gmeans  = (const float*)d_in[13];
  const float* glogvar = (const float*)d_in[14];
  const float* glogw   = (const float*)d_in[15];
  const float* pmeans  = (const float*)d_in[16];
  const float* plogvar = (const float*)d_in[17];
  const float* plogw   = (const float*)d_in[18];
  const int*   nullm   = (const int*)d_in[19];
  float* out = (float*)d_out;

  // workspace layout (needs ~153 MB): [eidx 4K][gidx 4K][alpha 580*1K][L^T 580*256K]
  char* ws = (char*)d_ws;
  int*   eidx = (int*)ws;
  int*   gidx = (int*)(ws + 4096);
  float* wsA  = (float*)(ws + 8192);
  float* wsU  = (float*)(ws + 8192 + (size_t)NGP * MPTS * sizeof(float));

  route_kernel<<<NTEST / 256, 256, 0, stream>>>(X_test, smean, sscale,
      gmeans, glogvar, glogw, pmeans, plogvar, plogw, eidx, gidx);

  gp_factor_kernel<<<NGP, 256, 0, stream>>>(X_exp, Y_exp, X_base, Y_base,
      lls_e, los_e, lno_e, lls_b, los_b, lno_b, wsU, wsA);

  predict_kernel<<<NTEST, 256, 0, stream>>>(X_test, X_exp, X_base,
      lls_e, los_e, lno_e, lls_b, los_b, lno_b, nullm, eidx, gidx, wsU, wsA, out);
}